// TAGLayer_39788577030304
// MI455X (gfx1250) — compile-verified
//
#include <hip/hip_runtime.h>
#include <math.h>

// Problem constants (from reference)
#define NN 64
#define CC 8
#define TT 300
#define VV 17
#define MM 12
#define HIPL 11
#define HIPR 12
#define TAUC 0.35f
#define EPSF 1e-6f
#define ROWLEN (VV*MM)        // 204 floats per (n,c,t) row

typedef __attribute__((ext_vector_type(2))) float v2f;
typedef __attribute__((ext_vector_type(4))) float v4f;
typedef __attribute__((ext_vector_type(8))) float v8f;

// ---------------------------------------------------------------------------
// Phase 1: per-sample graph construction. One block per n.
// Reduces x over T (pos) and T*V (ball score), then thread 0 builds the fused
// 12x12 adjacency exactly as the reference and writes it to workspace.
// ---------------------------------------------------------------------------
__global__ __launch_bounds__(256)
void tag_phase1(const float* __restrict__ x,
                const float* __restrict__ alpha_logits,
                float* __restrict__ adj /* NN * 144 floats */) {
    const int n = blockIdx.x;
    const int tid = threadIdx.x;

    __shared__ float posSum[3][MM];
    __shared__ float sSum[MM];
    if (tid < 36) posSum[tid / MM][tid % MM] = 0.0f;
    else if (tid < 48) sSum[tid - 36] = 0.0f;
    __syncthreads();

    float lp[3][MM];
    float ls[MM];
    for (int c = 0; c < 3; ++c)
        for (int m = 0; m < MM; ++m) lp[c][m] = 0.0f;
    for (int m = 0; m < MM; ++m) ls[m] = 0.0f;

    const size_t nbase = (size_t)n * CC * TT * VV * MM;
    for (int t = tid; t < TT; t += blockDim.x) {
        for (int c = 0; c < 3; ++c) {
            const float* px = x + nbase + ((size_t)c * TT + t) * ROWLEN;
            for (int m = 0; m < MM; ++m)
                lp[c][m] += 0.5f * (px[HIPL * MM + m] + px[HIPR * MM + m]);
        }
        const float* ps = x + nbase + ((size_t)3 * TT + t) * ROWLEN;
        for (int v = 0; v < VV; ++v)
            for (int m = 0; m < MM; ++m)
                ls[m] += ps[v * MM + m];
    }
    for (int c = 0; c < 3; ++c)
        for (int m = 0; m < MM; ++m) atomicAdd(&posSum[c][m], lp[c][m]);
    for (int m = 0; m < MM; ++m) atomicAdd(&sSum[m], ls[m]);
    __syncthreads();

    if (tid == 0) {
        float pos[MM][3];
        for (int m = 0; m < MM; ++m)
            for (int c = 0; c < 3; ++c) pos[m][c] = posSum[c][m] * (1.0f / TT);
        float s[MM];
        for (int m = 0; m < MM; ++m) s[m] = sSum[m] * (1.0f / (TT * VV));

        // pairwise euclidean distances among persons
        float d[MM][MM];
        for (int i = 0; i < MM; ++i)
            for (int jj = 0; jj < MM; ++jj) {
                float dx = pos[i][0] - pos[jj][0];
                float dy = pos[i][1] - pos[jj][1];
                float dz = pos[i][2] - pos[jj][2];
                d[i][jj] = sqrtf(dx * dx + dy * dy + dz * dz);
            }

        // alpha = softmax(alpha_logits)
        float al[4];
        float amax = -1e30f;
        for (int k = 0; k < 4; ++k) { al[k] = alpha_logits[k]; amax = fmaxf(amax, al[k]); }
        float asum = 0.0f;
        for (int k = 0; k < 4; ++k) { al[k] = expf(al[k] - amax); asum += al[k]; }
        for (int k = 0; k < 4; ++k) al[k] /= asum;

        float Aknn[MM][MM];
        for (int i = 0; i < MM; ++i)
            for (int jj = 0; jj < MM; ++jj) Aknn[i][jj] = 0.0f;

        const int klist[4] = {2, 3, 4, 5};
        for (int kk = 0; kk < 4; ++kk) {
            const int k = klist[kk];
            float Ak[MM][MM];
            for (int i = 0; i < MM; ++i)
                for (int jj = 0; jj < MM; ++jj) Ak[i][jj] = 0.0f;
            for (int i = 0; i < MM; ++i) {
                bool used[MM];
                for (int jj = 0; jj < MM; ++jj) used[jj] = false;
                for (int sel = 0; sel < k; ++sel) {  // lowest-index ties, like lax.top_k
                    int best = 0; float bv = 1e30f;
                    for (int jj = 0; jj < MM; ++jj)
                        if (!used[jj] && d[i][jj] < bv) { bv = d[i][jj]; best = jj; }
                    used[best] = true;
                    Ak[i][best] += 1.0f;
                }
                Ak[i][i] += 1.0f;  // + eye (diag -> 2, self always in kNN)
            }
            float r[MM];
            for (int i = 0; i < MM; ++i) {
                float deg = 0.0f;
                for (int jj = 0; jj < MM; ++jj) deg += Ak[i][jj];
                r[i] = 1.0f / sqrtf(deg + EPSF);
            }
            const float a = al[kk];
            for (int i = 0; i < MM; ++i)
                for (int jj = 0; jj < MM; ++jj)
                    Aknn[i][jj] += a * r[i] * Ak[i][jj] * r[jj];
        }

        // soft ball-star graph
        float zmax = -1e30f;
        for (int m = 0; m < MM; ++m) zmax = fmaxf(zmax, s[m] * (1.0f / TAUC));
        float p[MM]; float psum = 0.0f;
        for (int m = 0; m < MM; ++m) { p[m] = expf(s[m] * (1.0f / TAUC) - zmax); psum += p[m]; }
        float pmax = 0.0f;
        for (int m = 0; m < MM; ++m) { p[m] /= psum; pmax = fmaxf(pmax, p[m]); }

        float Ab[MM][MM];
        for (int i = 0; i < MM; ++i)
            for (int jj = 0; jj < MM; ++jj)
                Ab[i][jj] = p[i] + p[jj] + (i == jj ? 1.0f : 0.0f);
        float rb[MM];
        for (int i = 0; i < MM; ++i) {
            float deg = 0.0f;
            for (int jj = 0; jj < MM; ++jj) deg += Ab[i][jj];
            rb[i] = 1.0f / sqrtf(deg + EPSF);
        }

        float w = (pmax - 1.0f / MM) / (1.0f - 1.0f / MM + EPSF);
        w = fminf(fmaxf(w, 0.0f), 1.0f);

        float* outA = adj + (size_t)n * (MM * MM);
        for (int i = 0; i < MM; ++i)
            for (int jj = 0; jj < MM; ++jj)
                outA[i * MM + jj] = w * (rb[i] * Ab[i][jj] * rb[jj]) + (1.0f - w) * Aknn[i][jj];
    }
}

// ---------------------------------------------------------------------------
// Phase 2: one wave32 per (n, c, 16-row t-tile).
//  agg(16x12) = root(16x12) x Adj^T(12x12)  via 3x V_WMMA_F32_16X16X4_F32,
//  then a coalesced non-temporal float4 streaming pass
//  out = x + coef*(agg - root).
// ---------------------------------------------------------------------------
__global__ __launch_bounds__(32)
void tag_phase2(const float* __restrict__ x,
                const float* __restrict__ adj,
                const float* __restrict__ lambda_fuse,
                const float* __restrict__ tag_gate,
                float* __restrict__ out) {
    const int t0 = blockIdx.x * 16;
    const int c  = blockIdx.y;
    const int n  = blockIdx.z;
    const int lane = threadIdx.x;                // 0..31 (wave32)
    const int validRows = (TT - t0 < 16) ? (TT - t0) : 16;

    __shared__ alignas(16) float aggBuf[16 * MM];   // 192 floats
    __shared__ alignas(16) float rootBuf[16 * MM];  // 192 floats

    const float coef = tanhf(tag_gate[0]) * lambda_fuse[0];

    const size_t tileBase = (((size_t)n * CC + c) * TT + t0) * ROWLEN;
    const float* xt = x + tileBase;

    // --- B operand: B[k][u] = Adj[u][k]. ISA layout mirrors A:
    //     VGPR0: lanes0-15 K=0 / lanes16-31 K=2 ; VGPR1: K=1 / K=3.
    const float* An = adj + (size_t)n * (MM * MM);
    const int j    = lane & 15;                  // column u (person)
    const int koff = (lane < 16) ? 0 : 2;
    v2f B[3];
    for (int s = 0; s < 3; ++s) {
        float b0 = 0.0f, b1 = 0.0f;
        if (j < MM) {
            b0 = An[j * MM + 4 * s + koff];
            b1 = An[j * MM + 4 * s + koff + 1];
        }
        B[s].x = b0; B[s].y = b1;
    }

    // --- A operand: root rows. Lane L (and L+16) carry row L, K pairs.
    const int row = lane & 15;
    const bool rvalid = (row < validRows);
    v2f Aop[3];
    for (int s = 0; s < 3; ++s) {
        float a0 = 0.0f, a1 = 0.0f;
        if (rvalid) {
            const float* pr = xt + (size_t)row * ROWLEN;
            const int m = 4 * s + koff;
            a0 = 0.5f * (pr[HIPL * MM + m]     + pr[HIPR * MM + m]);
            a1 = 0.5f * (pr[HIPL * MM + m + 1] + pr[HIPR * MM + m + 1]);
        }
        Aop[s].x = a0; Aop[s].y = a1;
    }

    // --- three chained K=4 fp32 WMMAs (K total = 12)
    v8f acc = {};
    acc = __builtin_amdgcn_wmma_f32_16x16x4_f32(false, Aop[0], false, B[0], (short)0, acc, false, false);
    acc = __builtin_amdgcn_wmma_f32_16x16x4_f32(false, Aop[1], false, B[1], (short)0, acc, false, false);
    acc = __builtin_amdgcn_wmma_f32_16x16x4_f32(false, Aop[2], false, B[2], (short)0, acc, false, false);

    // --- spill D (agg) to LDS: VGPR r, lanes0-15 -> row r, lanes16-31 -> row r+8, col = j
    const int rbase = (lane < 16) ? 0 : 8;
    if (j < MM) {
        for (int r = 0; r < 8; ++r)
            aggBuf[(rbase + r) * MM + j] = acc[r];
    }
    // --- spill root (A layout) to LDS
    for (int s = 0; s < 3; ++s) {
        const int m = 4 * s + koff;
        rootBuf[row * MM + m]     = Aop[s].x;
        rootBuf[row * MM + m + 1] = Aop[s].y;
    }
    __syncthreads();

    // --- scaled delta, reuse aggBuf
    for (int idx = lane; idx < validRows * MM; idx += 32)
        aggBuf[idx] = coef * (aggBuf[idx] - rootBuf[idx]);
    __syncthreads();

    // --- streaming pass: the tile is validRows * 204 contiguous floats.
    //     204 = 4*51 so every float4 is inside one row, m0 in {0,4,8} -> aligned b128 LDS fetch.
    //     Zero reuse of this data -> non-temporal loads AND stores (don't churn
    //     the 192MB L2 / WGP$ with write-once, read-once streaming traffic).
    float* ot = out + tileBase;
    const int nF4 = validRows * (ROWLEN / 4);     // 51 per row
    const v4f* xv = reinterpret_cast<const v4f*>(xt);
    v4f*       ov = reinterpret_cast<v4f*>(ot);
    const v4f* dv = reinterpret_cast<const v4f*>(aggBuf);
    for (int i = lane; i < nF4; i += 32) {
        const int ofs = i * 4;
        const int r   = ofs / ROWLEN;
        const int rem = ofs - r * ROWLEN;
        const int m0  = rem % MM;                 // 0, 4 or 8
        const v4f xval = __builtin_nontemporal_load(&xv[i]);
        const v4f dval = dv[(r * MM + m0) >> 2];
        const v4f o = xval + dval;
        __builtin_nontemporal_store(o, &ov[i]);
    }
}

// ---------------------------------------------------------------------------
extern "C" void kernel_launch(void* const* d_in, const int* in_sizes, int n_in,
                              void* d_out, int out_size, void* d_ws, size_t ws_size,
                              hipStream_t stream) {
    const float* x            = (const float*)d_in[0];
    const float* alpha_logits = (const float*)d_in[1];
    const float* lambda_fuse  = (const float*)d_in[2];
    const float* tag_gate     = (const float*)d_in[3];
    float* out = (float*)d_out;
    float* adj = (float*)d_ws;   // NN * 144 floats = 36864 bytes

    tag_phase1<<<dim3(NN), dim3(256), 0, stream>>>(x, alpha_logits, adj);

    dim3 g2((TT + 15) / 16, CC, NN);  // 19 x 8 x 64 tiles, one wave32 each
    tag_phase2<<<g2, dim3(32), 0, stream>>>(x, adj, lambda_fuse, tag_gate, out);
}